// MultiTaskLoss_25898652795674
// MI455X (gfx1250) — compile-verified
//
#include <hip/hip_runtime.h>
#include <stdint.h>

// ---------------------------------------------------------------------------
// CTC loss (PyTorch semantics: blank=0, mean reduction, zero_infinity) on
// gfx1250. Latency-bound serial recurrence -> one wave32 per (task, sample),
// zero barriers (single-wave WG, LDS in-order within wave), deep (15-ahead)
// async global->LDS prefetch pipeline of per-timestep logits rows (ASYNCcnt),
// log-softmax denominators precomputed into LDS, and a reduced-TRANS lse3
// (exp-of-max == 1 identity). Both tasks fused into one launch so they run
// concurrently on different WGPs.
// ---------------------------------------------------------------------------

#define NEGV (-1e30f)

__device__ __forceinline__ void wait_async0() {
  asm volatile("s_wait_asynccnt 0x0" ::: "memory");
}
__device__ __forceinline__ void wait_async14() {   // allow 14 in flight
  asm volatile("s_wait_asynccnt 0xe" ::: "memory");
}

// Per-lane 16B async copy of one logits row (C floats) into LDS.
template<int C>
__device__ __forceinline__ void async_row_to_lds(const float* __restrict__ gsrc,
                                                 float* ldsdst, int lane) {
  constexpr int CHUNKS = (C * 4) / 16;   // 16 bytes per lane
  if (lane < CHUNKS) {
    unsigned laddr = (unsigned)(unsigned long long)(ldsdst + lane * 4);
    unsigned long long gaddr = (unsigned long long)(gsrc + lane * 4);
    // VDST = LDS byte address (low 32 bits of generic LDS pointer = LDS
    // offset, per flat->LDS truncation), VADDR = 64-bit global address.
    asm volatile("global_load_async_to_lds_b128 %0, %1, off"
                 :: "v"(laddr), "v"(gaddr)
                 : "memory");
  }
}

template<int C, int S, int NE>
__device__ __forceinline__ void run_ctc(const float* __restrict__ logits,
                                        const int*   __restrict__ targets,
                                        const int*   __restrict__ in_len,
                                        const int*   __restrict__ tg_len,
                                        float*       __restrict__ out_partial,
                                        int T, int Bn, int b, int lane) {
  constexpr int L    = 2 * S + 1;
  constexpr int NBUF = 16;           // prefetch ring
  constexpr int D    = 15;           // prefetch depth (<= NBUF-1)
  static_assert(32 * NE >= L, "NE too small");

  __shared__ float lse_buf[2048];                   // log-softmax denominators
  __shared__ float alpha[2][L + 2];                 // ping-pong + 2 NEG guards
  __shared__ __align__(16) float rowbuf[NBUF][C];   // async-prefetched rows

  int len = in_len[b];
  if (len > T) len = T;
  if (len < 1) len = 1;
  const int tl = tg_len[b];
  if (T > 2048) T = 2048;            // reference T=2000; guard LDS bounds

  const float* lg = logits + (size_t)b * T * C;

  // Guard slots (alpha[l-1], alpha[l-2] reads at l=0,1); never overwritten
  // (all recurrence writes go to index l+2 >= 2).
  if (lane < 2) { alpha[0][lane] = NEGV; alpha[1][lane] = NEGV; }

  // ---- Pre-pass: lse over C for every t (parallel over t across lanes).
  // Side effect: all rows become L2-resident for the recurrence prefetches.
  for (int t = lane; t < T; t += 32) {
    const float4* row = (const float4*)(lg + (size_t)t * C);
    float r[C];
    float mx = -3.0e38f;
#pragma unroll
    for (int c4 = 0; c4 < C / 4; ++c4) {
      float4 v = row[c4];
      r[4*c4+0] = v.x; r[4*c4+1] = v.y; r[4*c4+2] = v.z; r[4*c4+3] = v.w;
      mx = fmaxf(mx, fmaxf(fmaxf(v.x, v.y), fmaxf(v.z, v.w)));
    }
    float s = 0.f;
#pragma unroll
    for (int c = 0; c < C; ++c) s += __expf(r[c] - mx);
    lse_buf[t] = mx + __logf(s);
  }

  // ---- Extended targets + can_skip, held in registers ----
  int  e[NE];
  bool cs[NE];
  const int* tg = targets + (size_t)b * S;
#pragma unroll
  for (int k = 0; k < NE; ++k) {
    int l = lane + 32 * k;
    int ev = 0; bool c = false;
    if (l < L && (l & 1)) {
      ev = tg[l >> 1];
      if (l >= 2) {
        int pv = tg[(l >> 1) - 1];
        c = (ev != 0) && (ev != pv);
      }
    }
    e[k]  = ev;
    cs[k] = c;
  }

  // ---- Prologue: fill the prefetch pipeline with rows 0..D-1 ----
  {
    const int npre = (len < D) ? len : D;
    for (int i = 0; i < npre; ++i)
      async_row_to_lds<C>(lg + (size_t)i * C, rowbuf[i & (NBUF - 1)], lane);
  }

  // ---- t = 0: init alpha ----
  if (0 < len - D) { wait_async14(); async_row_to_lds<C>(lg + (size_t)D * C, rowbuf[D & (NBUF - 1)], lane); }
  else             { wait_async0(); }
  {
    const float lse0 = lse_buf[0];
#pragma unroll
    for (int k = 0; k < NE; ++k) {
      int l = lane + 32 * k;
      if (l < L) {
        float a = NEGV;
        if (l == 0)      a = rowbuf[0][0]    - lse0;   // blank
        else if (l == 1) a = rowbuf[0][e[k]] - lse0;   // first label
        alpha[0][l + 2] = a;
      }
    }
  }

  // ---- Serial recurrence (alpha frozen past len -> just stop there) ----
  for (int t = 1; t < len; ++t) {
    if (t < len - D) {
      wait_async14();                                  // oldest row (t) landed
      async_row_to_lds<C>(lg + (size_t)(t + D) * C,
                          rowbuf[(t + D) & (NBUF - 1)], lane);
    } else {
      wait_async0();                                   // tail: full drain
    }
    const int rb = t & (NBUF - 1);
    const float lse_t = lse_buf[t];
    const float* __restrict__ A  = alpha[(t - 1) & 1];
    float*       __restrict__ An = alpha[t & 1];
#pragma unroll
    for (int k = 0; k < NE; ++k) {
      int l = lane + 32 * k;
      if (l < L) {
        float a1 = A[l + 2];                      // alpha[l]
        float a2 = A[l + 1];                      // alpha[l-1]
        float a3 = cs[k] ? A[l] : NEGV;           // alpha[l-2] if can_skip
        // lse3 with exp(max-max)==1 exactly: 2 exps + 1 log (was 3 + 1).
        float p  = fminf(a1, a2);
        float q  = fmaxf(a1, a2);
        float m  = fmaxf(q, a3);
        float x  = fminf(q, a3);
        float ss = 1.0f + __expf(x - m) + __expf(p - m);
        float lp = rowbuf[rb][e[k]] - lse_t;      // logp[b,t,ext[l]]
        An[l + 2] = lp + m + __logf(ss);
      }
    }
  }

  // ---- Finalize: loss_b = -logaddexp(alpha[2tl-1], alpha[2tl]) ----
  if (lane == 0) {
    const float* Af = alpha[(len - 1) & 1];
    float a1 = Af[(2 * tl - 1) + 2];
    float a2 = Af[(2 * tl) + 2];
    float m  = fmaxf(a1, a2);
    float mn = fminf(a1, a2);
    float loss = -(m + __logf(1.0f + __expf(mn - m)));
    if (loss > 1e29f) loss = 0.f;                 // zero_infinity
    out_partial[b] = loss / (float)tl / (float)Bn;
  }
}

// One launch, both tasks: blocks [0,Bn) = error task, [Bn,2Bn) = phoneme task.
__global__ void __launch_bounds__(32)
ctc_fused_kernel(const float* __restrict__ err_logits,
                 const float* __restrict__ ph_logits,
                 const int*   __restrict__ err_tg,
                 const int*   __restrict__ ph_tg,
                 const int*   __restrict__ err_il,
                 const int*   __restrict__ ph_il,
                 const int*   __restrict__ err_tl,
                 const int*   __restrict__ ph_tl,
                 float*       __restrict__ out_partial,
                 int T, int Bn) {
  const int lane = threadIdx.x;
  if ((int)blockIdx.x < Bn) {
    run_ctc<4, 50, 4>(err_logits, err_tg, err_il, err_tl,
                      out_partial, T, Bn, blockIdx.x, lane);
  } else {
    run_ctc<64, 200, 13>(ph_logits, ph_tg, ph_il, ph_tl,
                         out_partial + Bn, T, Bn, blockIdx.x - Bn, lane);
  }
}

// Deterministic fixed-order reduction of the 2*B per-sample contributions.
__global__ void reduce_sum_kernel(const float* __restrict__ part, int n,
                                  float* __restrict__ out) {
  if (threadIdx.x == 0 && blockIdx.x == 0) {
    float s = 0.f;
    for (int i = 0; i < n; ++i) s += part[i];
    out[0] = s;
  }
}

extern "C" void kernel_launch(void* const* d_in, const int* in_sizes, int n_in,
                              void* d_out, int out_size, void* d_ws, size_t ws_size,
                              hipStream_t stream) {
  const float* err_logits = (const float*)d_in[0];  // [B,T,4]
  const float* ph_logits  = (const float*)d_in[1];  // [B,T,64]
  const int*   err_tg     = (const int*)d_in[2];    // [B,50]
  const int*   ph_tg      = (const int*)d_in[3];    // [B,200]
  const int*   err_il     = (const int*)d_in[4];    // [B]
  const int*   ph_il      = (const int*)d_in[5];    // [B]
  const int*   err_tl     = (const int*)d_in[6];    // [B]
  const int*   ph_tl      = (const int*)d_in[7];    // [B]

  const int Bn = in_sizes[4];                 // 32
  const int T  = in_sizes[0] / (Bn * 4);      // 2000

  float* part = (float*)d_ws;                 // 2*B floats of scratch

  ctc_fused_kernel<<<2 * Bn, 32, 0, stream>>>(
      err_logits, ph_logits, err_tg, ph_tg,
      err_il, ph_il, err_tl, ph_tl, part, T, Bn);

  reduce_sum_kernel<<<1, 32, 0, stream>>>(part, 2 * Bn, (float*)d_out);
}